// ClippedRunningStats_6811818131595
// MI455X (gfx1250) — compile-verified
//
#include <hip/hip_runtime.h>
#include <math.h>

typedef float v2f __attribute__((ext_vector_type(2)));
typedef float v8f __attribute__((ext_vector_type(8)));

#define SHIFT_V 1000.0f
#define SAT_V   10000.0f
#define CCH     12
#define BATCH   32
#define HW      65536                 // 256*256 contiguous per (b,c) slab
#define BPS     8                     // blocks per slab
#define EPB     (HW / BPS)            // 8192 elements per block
#define THREADS 256
#define ITERS   (EPB / (THREADS * 4)) // 8 float4 iterations per thread

// ---- monotone ordering of float bits for u32 atomic min/max ----
__device__ __forceinline__ unsigned f2ord(float f) {
    unsigned u = __float_as_uint(f);
    return (u & 0x80000000u) ? ~u : (u | 0x80000000u);
}
__device__ __forceinline__ float ord2f(unsigned o) {
    unsigned u = (o & 0x80000000u) ? (o & 0x7FFFFFFFu) : ~o;
    return __uint_as_float(u);
}

// ws layout (12 entries each): [0]=sum(f32) [1]=sumsq(f32) [2]=count(f32)
//                              [3]=min(ordered u32) [4]=max(ordered u32)
__global__ void crs_init_ws(unsigned* ws) {
    int t = threadIdx.x;
    if (t < CCH) {
        float* wf = (float*)ws;
        wf[0 * CCH + t] = 0.0f;
        wf[1 * CCH + t] = 0.0f;
        wf[2 * CCH + t] = 0.0f;
        ws[3 * CCH + t] = f2ord(INFINITY);    // min identity
        ws[4 * CCH + t] = f2ord(-INFINITY);   // max identity
    }
}

__global__ __launch_bounds__(THREADS) void crs_stats_kernel(
    const float* __restrict__ x, unsigned* __restrict__ ws) {
    const int slab = blockIdx.x / BPS;       // (b*C + c), 0..383
    const int part = blockIdx.x % BPS;
    const int ch   = slab % CCH;
    const int tid  = threadIdx.x;
    const float4* p =
        (const float4*)(x + (size_t)slab * HW + (size_t)part * EPB);

    v2f ones; ones.x = 1.0f; ones.y = 1.0f;
    v8f sumD = {};   // WMMA accumulators: D = A*ones + D  (row-sum engine)
    v8f sqD  = {};
    v8f cntD = {};
    float mn = INFINITY, mx = -INFINITY;

    #pragma unroll
    for (int i = 0; i < ITERS; ++i) {
        float4 v = p[i * THREADS + tid];
        float e0 = v.x, e1 = v.y, e2 = v.z, e3 = v.w;
        float xf[4], sq[4], vf[4];
        float ein[4] = {e0, e1, e2, e3};
        #pragma unroll
        for (int j = 0; j < 4; ++j) {
            float xs    = ein[j] + SHIFT_V;
            bool  valid = xs > 0.0f;
            float c     = fminf(fmaxf(xs, 0.0f), SAT_V);  // invalid -> exactly 0
            xf[j] = c;
            sq[j] = c * c;
            vf[j] = valid ? 1.0f : 0.0f;
            mn = fminf(mn, valid ? c : INFINITY);
            mx = fmaxf(mx, c);   // safe unmasked: valid values > 0, invalid give 0
        }
        v2f a;
        // two K-slices per iteration per stat; independent chains pipeline on XDL
        a.x = xf[0]; a.y = xf[1];
        sumD = __builtin_amdgcn_wmma_f32_16x16x4_f32(false, a, false, ones, (short)0, sumD, false, false);
        a.x = sq[0]; a.y = sq[1];
        sqD  = __builtin_amdgcn_wmma_f32_16x16x4_f32(false, a, false, ones, (short)0, sqD,  false, false);
        a.x = vf[0]; a.y = vf[1];
        cntD = __builtin_amdgcn_wmma_f32_16x16x4_f32(false, a, false, ones, (short)0, cntD, false, false);
        a.x = xf[2]; a.y = xf[3];
        sumD = __builtin_amdgcn_wmma_f32_16x16x4_f32(false, a, false, ones, (short)0, sumD, false, false);
        a.x = sq[2]; a.y = sq[3];
        sqD  = __builtin_amdgcn_wmma_f32_16x16x4_f32(false, a, false, ones, (short)0, sqD,  false, false);
        a.x = vf[2]; a.y = vf[3];
        cntD = __builtin_amdgcn_wmma_f32_16x16x4_f32(false, a, false, ones, (short)0, cntD, false, false);
    }

    // Collapse D: lane partial = sum of 8 D VGPRs (rows 0-7 in lanes<16, rows 8-15 in lanes>=16)
    float ps = 0.f, pq = 0.f, pc = 0.f;
    #pragma unroll
    for (int j = 0; j < 8; ++j) { ps += sumD[j]; pq += sqD[j]; pc += cntD[j]; }
    ps += __shfl_xor(ps, 16, 32);
    pq += __shfl_xor(pq, 16, 32);
    pc += __shfl_xor(pc, 16, 32);
    #pragma unroll
    for (int m = 1; m < 32; m <<= 1) {
        mn = fminf(mn, __shfl_xor(mn, m, 32));
        mx = fmaxf(mx, __shfl_xor(mx, m, 32));
    }

    __shared__ float shs[THREADS / 32], shq[THREADS / 32], shc[THREADS / 32],
                     shmn[THREADS / 32], shmx[THREADS / 32];
    const int lane = tid & 31, w = tid >> 5;
    if (lane == 0) { shs[w] = ps; shq[w] = pq; shc[w] = pc; shmn[w] = mn; shmx[w] = mx; }
    __syncthreads();
    if (tid == 0) {
        float bs = 0.f, bq = 0.f, bc = 0.f, bmn = INFINITY, bmx = -INFINITY;
        #pragma unroll
        for (int k = 0; k < THREADS / 32; ++k) {
            bs += shs[k]; bq += shq[k]; bc += shc[k];
            bmn = fminf(bmn, shmn[k]); bmx = fmaxf(bmx, shmx[k]);
        }
        float* wf = (float*)ws;
        atomicAdd(&wf[0 * CCH + ch], bs);
        atomicAdd(&wf[1 * CCH + ch], bq);
        atomicAdd(&wf[2 * CCH + ch], bc);
        atomicMin(&ws[3 * CCH + ch], f2ord(bmn));
        atomicMax(&ws[4 * CCH + ch], f2ord(bmx));
    }
}

__global__ void crs_finalize(const unsigned* __restrict__ ws,
                             const float* __restrict__ mean, const float* __restrict__ var,
                             const float* __restrict__ stdv, const float* __restrict__ count,
                             const float* __restrict__ minv, const float* __restrict__ maxv,
                             float* __restrict__ out) {
    int c = threadIdx.x;
    if (c >= CCH) return;
    const float* wf = (const float*)ws;
    float s    = wf[0 * CCH + c];
    float ssum = wf[1 * CCH + c];
    float n    = wf[2 * CCH + c];
    float bmin = ord2f(ws[3 * CCH + c]);
    float bmax = ord2f(ws[4 * CCH + c]);
    if (!(n > 0.0f)) { bmin = INFINITY; bmax = -INFINITY; }

    float n_safe = fmaxf(n, 1.0f);
    float bmean  = (n > 0.0f) ? (s / n_safe) : 0.0f;
    float bvar   = fmaxf(ssum / n_safe - bmean * bmean, 0.0f);
    bvar = (n > 1.0f) ? bvar : 0.0f;

    float m = mean[c], v = var[c], sd = stdv[c], cnt = count[c];
    float delta = bmean - m;
    float total = cnt + n;
    float denom = fmaxf(total, 1.0f);
    float merged_mean = m + delta * (n / denom);
    float M2 = v * cnt + bvar * n + delta * delta * cnt * n / denom;

    bool  upd  = n > 0.0f;
    float nm   = upd ? merged_mean : m;
    float nv   = upd ? (M2 / denom) : v;
    float nsd  = upd ? sqrtf(nv + 1e-8f) : sd;
    float ncnt = cnt + n;
    float nmin = fminf(minv[c], bmin);
    float nmax = fmaxf(maxv[c], bmax);

    out[0 * CCH + c] = nm;
    out[1 * CCH + c] = nv;
    out[2 * CCH + c] = nsd;
    out[3 * CCH + c] = ncnt;
    out[4 * CCH + c] = nmin;
    out[5 * CCH + c] = nmax;
}

extern "C" void kernel_launch(void* const* d_in, const int* in_sizes, int n_in,
                              void* d_out, int out_size, void* d_ws, size_t ws_size,
                              hipStream_t stream) {
    const float* x     = (const float*)d_in[0];
    const float* mean  = (const float*)d_in[1];
    const float* var   = (const float*)d_in[2];
    const float* stdv  = (const float*)d_in[3];
    const float* count = (const float*)d_in[4];
    const float* minv  = (const float*)d_in[5];
    const float* maxv  = (const float*)d_in[6];
    float*    out = (float*)d_out;
    unsigned* ws  = (unsigned*)d_ws;

    crs_init_ws<<<1, 64, 0, stream>>>(ws);
    crs_stats_kernel<<<BATCH * CCH * BPS, THREADS, 0, stream>>>(x, ws);
    crs_finalize<<<1, 32, 0, stream>>>(ws, mean, var, stdv, count, minv, maxv, out);
}